// GNN_Encoder_7851200218009
// MI455X (gfx1250) — compile-verified
//
#include <hip/hip_runtime.h>

typedef float v2f __attribute__((ext_vector_type(2)));
typedef float v8f __attribute__((ext_vector_type(8)));

#define NEG_SLOPE 0.3f

// ---------------------------------------------------------------- init
__global__ void k_init(float* __restrict__ aggbuf, float* __restrict__ dis,
                       float* __restrict__ pcnt, float* __restrict__ pool,
                       long n64, int N) {
  long i = (long)blockIdx.x * blockDim.x + threadIdx.x;
  if (i < n64) aggbuf[i] = 0.0f;
  if (i < N) dis[i] = 0.0f;
  if (i < 64) pcnt[i] = 0.0f;
  if (i < 64 * 32) pool[i] = 0.0f;
}

// ------------------------------------------------------- degree (atomics)
__global__ void k_deg(const int* __restrict__ dst, float* __restrict__ deg, int E) {
  int e = blockIdx.x * blockDim.x + threadIdx.x;
  if (e < E) unsafeAtomicAdd(&deg[dst[e]], 1.0f);
}

// deg -> dis = rsqrt(deg + 1)   (self-loop adds 1, so deg >= 1 always)
__global__ void k_rsqrt(float* __restrict__ dis, int N) {
  int i = blockIdx.x * blockDim.x + threadIdx.x;
  if (i < N) dis[i] = rsqrtf(dis[i] + 1.0f);
}

// -------------------------------------------- edge aggregation (wave/edge)
// out[dst] += dis[src]*dis[dst] * hin[src]   for all E edges (64 floats each)
__global__ void k_agg(const int* __restrict__ src, const int* __restrict__ dst,
                      const float* __restrict__ dis, const float* __restrict__ hin,
                      float* __restrict__ out, int E) {
  int t = blockIdx.x * blockDim.x + threadIdx.x;
  int e = t >> 5;
  int lane = t & 31;
  if (e >= E) return;
  int s = src[e];
  int d = dst[e];
  float nm = dis[s] * dis[d];
  const float* row = hin + (size_t)s * 64 + lane * 2;
  float vx = row[0];
  float vy = row[1];
  float* orow = out + (size_t)d * 64 + lane * 2;
  unsafeAtomicAdd(orow, nm * vx);
  unsafeAtomicAdd(orow + 1, nm * vy);
}

// ------------------------------------------------- fused WMMA GEMM layers
// A-row m = agg[m] + dis[m]^2 * hsrc[m]  (adds self-loop term)
// D = A @ W  via V_WMMA_F32_16X16X4_F32; one wave owns rows [m0, m0+16).
// MODE 0: out = leaky(D)              (ld 64), zero aggz tile
// MODE 1: out = relu(leaky(D))=max(D,0) (ld 64), zero aggz tile
// MODE 2: out = D + bias              (ld 32)
// NOTE: agg/hsrc/out/aggz may alias each other (in-place layers) -> no restrict.
template <int LDW, int NB, int MODE>
__global__ void k_gemm(const float* agg, const float* hsrc,
                       const float* __restrict__ dis,
                       const float* __restrict__ W,
                       const float* __restrict__ bias,
                       float* out, float* aggz, int N) {
  int t = blockIdx.x * blockDim.x + threadIdx.x;
  int wv = t >> 5;
  int lane = t & 31;
  int m0 = wv << 4;
  if (m0 >= N) return;
  int half = lane >> 4;   // A layout: lanes 0-15 K pair {0,1}; 16-31 pair {2,3}
  int l15 = lane & 15;
  int m = m0 + l15;
  int mc = m < N ? m : N - 1;
  float dv = dis[mc];
  float d2 = dv * dv;

  const float* ar = agg + (size_t)mc * 64;
  const float* hr = hsrc + (size_t)mc * 64;
  v2f a[16];
#pragma unroll
  for (int ks = 0; ks < 16; ++ks) {
    int k = ks * 4 + half * 2;
    a[ks][0] = ar[k] + d2 * hr[k];
    a[ks][1] = ar[k + 1] + d2 * hr[k + 1];
  }

  // Per-lane W base; all K offsets become compile-time immediates.
  const float* wb0 = W + half * 2 * LDW + l15;
#pragma unroll
  for (int nb = 0; nb < NB; ++nb) {
    const float* wb = wb0 + nb * 16;
    v2f b[16];
#pragma unroll
    for (int ks = 0; ks < 16; ++ks) {      // batch all B loads first
      b[ks][0] = wb[ks * 4 * LDW];
      b[ks][1] = wb[ks * 4 * LDW + LDW];
    }
    v8f acc = {};
#pragma unroll
    for (int ks = 0; ks < 16; ++ks) {      // back-to-back WMMA chain
      acc = __builtin_amdgcn_wmma_f32_16x16x4_f32(
          false, a[ks], false, b[ks], (short)0, acc, false, false);
    }
    int n = nb * 16 + l15;
#pragma unroll
    for (int r = 0; r < 8; ++r) {
      int row = m0 + r + half * 8;         // D layout: vgpr r -> rows r / r+8
      if (row >= N) continue;
      float v = acc[r];
      if (MODE == 0) {
        v = v > 0.0f ? v : NEG_SLOPE * v;
        out[(size_t)row * 64 + n] = v;
        aggz[(size_t)row * 64 + n] = 0.0f;
      } else if (MODE == 1) {
        v = fmaxf(v, 0.0f);                // relu(leaky(x)) == max(x,0)
        out[(size_t)row * 64 + n] = v;
        aggz[(size_t)row * 64 + n] = 0.0f;
      } else {
        out[(size_t)row * 32 + n] = v + bias[n];
      }
    }
  }
}

// -------------------------------------------------------- scatter-mean pool
__global__ void k_pool(const float* __restrict__ mean, const int* __restrict__ gidx,
                       float* __restrict__ psum, float* __restrict__ pcnt, int N) {
  int t = blockIdx.x * blockDim.x + threadIdx.x;
  int node = t >> 5;
  int lane = t & 31;  // D_Z == 32 == wave width
  if (node >= N) return;
  int g = gidx[node];
  unsafeAtomicAdd(&psum[g * 32 + lane], mean[(size_t)node * 32 + lane]);
  if (lane == 0) unsafeAtomicAdd(&pcnt[g], 1.0f);
}

__global__ void k_pooldiv(float* __restrict__ psum, const float* __restrict__ pcnt) {
  int t = blockIdx.x * blockDim.x + threadIdx.x;
  if (t < 64 * 32) psum[t] /= fmaxf(pcnt[t >> 5], 1.0f);
}

// ---------------------------------------------------------------- launcher
extern "C" void kernel_launch(void* const* d_in, const int* in_sizes, int n_in,
                              void* d_out, int out_size, void* d_ws, size_t ws_size,
                              hipStream_t stream) {
  const float* x  = (const float*)d_in[0];
  const int* ei   = (const int*)d_in[1];
  const int* gidx = (const int*)d_in[2];
  const float* W1 = (const float*)d_in[3];
  const float* W2 = (const float*)d_in[4];
  const float* Wm = (const float*)d_in[5];
  const float* bm = (const float*)d_in[6];
  const float* Wv = (const float*)d_in[7];
  const float* bv = (const float*)d_in[8];

  int N = in_sizes[0] / 64;   // 100000
  int E = in_sizes[1] / 2;    // 1600000
  const int* src = ei;
  const int* dst = ei + E;

  float* buf1 = (float*)d_ws;                 // h  (N x 64)
  float* buf2 = buf1 + (size_t)N * 64;        // aggregation accumulator (N x 64)
  float* dis  = buf2 + (size_t)N * 64;        // N
  float* pcnt = dis + N;                      // 64

  float* mean = (float*)d_out;                // N x 32
  float* lvar = mean + (size_t)N * 32;        // N x 32
  float* pool = mean + (size_t)2 * N * 32;    // 64 x 32

  const int B = 256;
  long n64 = (long)N * 64;
  long eth = (long)E * 32;            // threads for edge kernels (wave/edge)
  int gth = ((N + 15) / 16) * 32;     // threads for gemm (wave per 16 rows)
  long pth = (long)N * 32;            // threads for pooling

  k_init<<<(int)((n64 + B - 1) / B), B, 0, stream>>>(buf2, dis, pcnt, pool, n64, N);
  k_deg<<<(E + B - 1) / B, B, 0, stream>>>(dst, dis, E);
  k_rsqrt<<<(N + B - 1) / B, B, 0, stream>>>(dis, N);

  // layer 1: h1 = leaky((A @ x) @ W1)
  k_agg<<<(int)((eth + B - 1) / B), B, 0, stream>>>(src, dst, dis, x, buf2, E);
  k_gemm<64, 4, 0><<<(gth + B - 1) / B, B, 0, stream>>>(buf2, x, dis, W1, nullptr,
                                                        buf1, buf2, N);
  // layer 2: h = relu(leaky((A @ h1) @ W2))   (in-place on buf1)
  k_agg<<<(int)((eth + B - 1) / B), B, 0, stream>>>(src, dst, dis, buf1, buf2, E);
  k_gemm<64, 4, 1><<<(gth + B - 1) / B, B, 0, stream>>>(buf2, buf1, dis, W2, nullptr,
                                                        buf1, buf2, N);
  // shared aggregation a = A @ h, then mean = a@Wm+bm, logvar = a@Wv+bv
  k_agg<<<(int)((eth + B - 1) / B), B, 0, stream>>>(src, dst, dis, buf1, buf2, E);
  k_gemm<32, 2, 2><<<(gth + B - 1) / B, B, 0, stream>>>(buf2, buf1, dis, Wm, bm,
                                                        mean, nullptr, N);
  k_gemm<32, 2, 2><<<(gth + B - 1) / B, B, 0, stream>>>(buf2, buf1, dis, Wv, bv,
                                                        lvar, nullptr, N);
  // scatter-mean pooling of mean over graph_index
  k_pool<<<(int)((pth + B - 1) / B), B, 0, stream>>>(mean, gidx, pool, pcnt, N);
  k_pooldiv<<<(64 * 32 + B - 1) / B, B, 0, stream>>>(pool, pcnt);
}